// IntrinsicGrowthController_5729486372963
// MI455X (gfx1250) — compile-verified
//
#include <hip/hip_runtime.h>
#include <math.h>

typedef __attribute__((ext_vector_type(2))) float v2f;
typedef __attribute__((ext_vector_type(4))) float v4f;
typedef __attribute__((ext_vector_type(8))) float v8f;

#define B_DIM 16384
#define D_DIM 2048
#define H2    1024

__device__ __forceinline__ float wave_sum(float v) {
#pragma unroll
    for (int off = 16; off > 0; off >>= 1) v += __shfl_down(v, off, 32);
    return v;  // valid in lane 0
}

// ---------------------------------------------------------------------------
// Kernel 1: bandwidth-bound reduction over x/out/noise (402 MB streamed once).
// Non-temporal b128 loads for the streamed arrays (bigger than L2, zero reuse);
// input_mean (8 KB, reused 16K times) stays on the regular cached path.
// 1024 blocks x 256 threads; each wave owns 2 full rows (32 lanes x 16 float4).
// Writes 5 partials per block: [q*1024 + blk], q in {noise2, nov, abs, pe, pe2}
// ---------------------------------------------------------------------------
__global__ void igc_reduce(const v4f* __restrict__ x4,
                           const v4f* __restrict__ o4,
                           const v4f* __restrict__ n4,
                           const v4f* __restrict__ im4,
                           float* __restrict__ partials) {
    const int lane = threadIdx.x & 31;
    const int wave = threadIdx.x >> 5;
    const int blk  = blockIdx.x;
    const float invD = 1.0f / (float)D_DIM;

    float accN = 0.f, accV = 0.f, accA = 0.f, accP = 0.f, accP2 = 0.f;

    for (int r = 0; r < 2; ++r) {
        const int row = blk * 16 + wave * 2 + r;
        const size_t base = (size_t)row * (D_DIM / 4);
        float n2 = 0.f, nv = 0.f, pe = 0.f, ab = 0.f;
#pragma unroll 4
        for (int i = 0; i < 16; ++i) {
            const int c = i * 32 + lane;
            v4f xv = __builtin_nontemporal_load(&x4[base + c]);
            v4f ov = __builtin_nontemporal_load(&o4[base + c]);
            v4f zv = __builtin_nontemporal_load(&n4[base + c]);
            v4f iv = im4[c];
#pragma unroll
            for (int e = 0; e < 4; ++e) {
                n2 += zv[e] * zv[e];
                float dx = xv[e] - iv[e];
                nv += dx * dx;
                float dp = ov[e] - xv[e];
                pe += dp * dp;
                ab += fabsf(ov[e]);
            }
        }
        n2 = wave_sum(n2);
        nv = wave_sum(nv);
        pe = wave_sum(pe);
        ab = wave_sum(ab);
        if (lane == 0) {
            accN += n2; accV += nv; accA += ab;
            float per = pe * invD;
            accP += per; accP2 += per * per;
        }
    }

    __shared__ float s[8][5];
    if (lane == 0) {
        s[wave][0] = accN; s[wave][1] = accV; s[wave][2] = accA;
        s[wave][3] = accP; s[wave][4] = accP2;
    }
    __syncthreads();
    if (threadIdx.x == 0) {
        float t0 = 0.f, t1 = 0.f, t2 = 0.f, t3 = 0.f, t4 = 0.f;
        for (int w = 0; w < 8; ++w) {
            t0 += s[w][0]; t1 += s[w][1]; t2 += s[w][2]; t3 += s[w][3]; t4 += s[w][4];
        }
        partials[0 * 1024 + blk] = t0;
        partials[1 * 1024 + blk] = t1;
        partials[2 * 1024 + blk] = t2;
        partials[3 * 1024 + blk] = t3;
        partials[4 * 1024 + blk] = t4;
    }
}

// ---------------------------------------------------------------------------
// Kernel 2 (1 block): fold partials, usage stats, build sig[15], h = relu(sig@W1+b1)
// ---------------------------------------------------------------------------
__global__ void igc_finalize(const float* __restrict__ partials,
                             const float* __restrict__ usage,
                             const float* __restrict__ stats,
                             const float* __restrict__ gsig,
                             const float* __restrict__ rma_p,
                             const float* __restrict__ alpha_p,
                             const float* __restrict__ W1,
                             const float* __restrict__ b1,
                             float* __restrict__ h_out) {
    __shared__ float wpart[5][8];
    __shared__ float sig[16];
    const int lane = threadIdx.x & 31;
    const int wave = threadIdx.x >> 5;

    for (int q = 0; q < 5; ++q) {
        float v = 0.f;
        for (int i = threadIdx.x; i < 1024; i += 256) v += partials[q * 1024 + i];
        v = wave_sum(v);
        if (lane == 0) wpart[q][wave] = v;
    }
    __syncthreads();

    if (threadIdx.x == 0) {
        float red[5];
        for (int q = 0; q < 5; ++q) {
            float t = 0.f;
            for (int w = 0; w < 8; ++w) t += wpart[q][w];
            red[q] = t;
        }
        // operator_usage stats (64 elements)
        float sum_u = 0.f, max_u = -INFINITY, used = 0.f;
        for (int i = 0; i < 64; ++i) {
            float u = usage[i];
            sum_u += u;
            max_u = fmaxf(max_u, u);
            used += (u > 0.f) ? 1.f : 0.f;
        }
        float mean_u = sum_u * (1.0f / 64.0f);
        float inv_s = 1.0f / (sum_u + 1e-6f);
        float var = 0.f, ent = 0.f;
        for (int i = 0; i < 64; ++i) {
            float u = usage[i];
            float d = u - mean_u;
            var += d * d;
            float p = u * inv_s;
            ent -= p * logf(fmaxf(p, 1e-6f));
        }
        float std_u = sqrtf(var * (1.0f / 63.0f));  // ddof=1

        const float invBD = 1.0f / ((float)B_DIM * (float)D_DIM);
        const float invB  = 1.0f / (float)B_DIM;
        float rma   = rma_p[0];
        float alpha = alpha_p[0];
        float mean_pe = red[3] * invB;
        float new_avg = 0.99f * rma + 0.01f * mean_pe;
        float rvar = red[4] * invB - 2.0f * new_avg * mean_pe + new_avg * new_avg;

        float sg[15];
        sg[0]  = 1e-4f * red[0] * invBD;     // plasticity
        sg[1]  = red[1] * invBD;             // novelty
        sg[2]  = mean_pe;                    // prediction error
        sg[3]  = ent;                        // usage entropy
        sg[4]  = red[2] * invBD;             // sparsity
        sg[5]  = rma - mean_pe;              // reward delta (old avg)
        sg[6]  = rvar;                       // reward var (new avg)
        sg[7]  = mean_u;
        sg[8]  = max_u;
        sg[9]  = std_u;
        sg[10] = used * (1.0f / 64.0f);
        sg[11] = stats[0]; sg[12] = stats[1]; sg[13] = stats[2]; sg[14] = stats[3];
        for (int i = 0; i < 15; ++i) sig[i] = sg[i] + alpha * gsig[i];
    }
    __syncthreads();

    float sl[15];
#pragma unroll
    for (int k = 0; k < 15; ++k) sl[k] = sig[k];
    for (int j = threadIdx.x; j < D_DIM; j += 256) {
        float acc = b1[j];
#pragma unroll
        for (int k = 0; k < 15; ++k) acc = fmaf(sl[k], W1[k * D_DIM + j], acc);
        h_out[j] = fmaxf(acc, 0.f);
    }
}

// ---------------------------------------------------------------------------
// Kernel 3: GEMV heads via V_WMMA_F32_16X16X4_F32.
// 16 blocks x 8 waves = 128 waves; wave gw: head m = gw>>6, n-tile = gw&63.
// A rows broadcast h (rows of D identical; we read row 0), B = W[k][nb+nn]
// per ISA f32 layout: lane half kh = lane>>4, v0 holds K=2kh, v1 holds K=2kh+1
// for both A and B operands.
// ---------------------------------------------------------------------------
__global__ void igc_gemv_wmma(const float* __restrict__ h,
                              const float* __restrict__ Wg1, const float* __restrict__ bg1,
                              const float* __restrict__ Wg2,
                              const float* __restrict__ Wp1, const float* __restrict__ bp1,
                              const float* __restrict__ Wp2,
                              float* __restrict__ partial_out) {
    __shared__ float hs[D_DIM];
    {
        v4f* hs4 = (v4f*)hs;
        const v4f* h4 = (const v4f*)h;
        for (int i = threadIdx.x; i < D_DIM / 4; i += 256) hs4[i] = h4[i];
    }
    __syncthreads();

    const int lane = threadIdx.x & 31;
    const int wave = threadIdx.x >> 5;
    const int gw   = blockIdx.x * 8 + wave;    // 0..127
    const int m    = gw >> 6;                  // 0 = grow, 1 = prune
    const int tile = gw & 63;                  // 0..63

    const float* __restrict__ W  = m ? Wp1 : Wg1;
    const float* __restrict__ bb = m ? bp1 : bg1;
    const float* __restrict__ W2 = m ? Wp2 : Wg2;

    const int nb = tile * 16;
    const int nn = lane & 15;
    const int kh = lane >> 4;
    const float* __restrict__ wcol = W + nb + nn;

    v8f c = {0.f, 0.f, 0.f, 0.f, 0.f, 0.f, 0.f, 0.f};
    for (int k0 = 0; k0 < D_DIM; k0 += 4) {
        const int kk = k0 + 2 * kh;
        v2f a, b;
        a[0] = hs[kk];
        a[1] = hs[kk + 1];
        b[0] = wcol[(size_t)kk * H2];
        b[1] = wcol[(size_t)(kk + 1) * H2];
        c = __builtin_amdgcn_wmma_f32_16x16x4_f32(
            /*neg_a=*/false, a, /*neg_b=*/false, b,
            /*c_mod=*/(short)0, c, /*reuse_a=*/false, /*reuse_b=*/false);
    }

    float res = 0.f;
    if (lane < 16) {
        float u = fmaxf(c[0] + bb[nb + nn], 0.f);  // D row 0: VGPR0, lanes 0-15
        res = u * W2[nb + nn];
    }
    res = wave_sum(res);
    if (lane == 0) partial_out[gw] = res;
}

// ---------------------------------------------------------------------------
// Kernel 4: ordered final sum + sigmoid.
// ---------------------------------------------------------------------------
__global__ void igc_out(const float* __restrict__ partial_out,
                        const float* __restrict__ bg2,
                        const float* __restrict__ bp2,
                        float* __restrict__ out) {
    const int m = threadIdx.x;
    if (m < 2) {
        float s = 0.f;
        for (int i = 0; i < 64; ++i) s += partial_out[m * 64 + i];
        s += (m == 0) ? bg2[0] : bp2[0];
        out[m] = 1.0f / (1.0f + expf(-s));
    }
}

extern "C" void kernel_launch(void* const* d_in, const int* in_sizes, int n_in,
                              void* d_out, int out_size, void* d_ws, size_t ws_size,
                              hipStream_t stream) {
    const float* x     = (const float*)d_in[0];
    const float* out   = (const float*)d_in[1];
    const float* noise = (const float*)d_in[2];
    const float* usage = (const float*)d_in[3];
    const float* imean = (const float*)d_in[4];
    const float* rma   = (const float*)d_in[5];
    const float* stats = (const float*)d_in[6];
    const float* gsig  = (const float*)d_in[7];
    const float* W1    = (const float*)d_in[8];
    const float* b1    = (const float*)d_in[9];
    const float* Wg1   = (const float*)d_in[10];
    const float* bg1   = (const float*)d_in[11];
    const float* Wg2   = (const float*)d_in[12];
    const float* bg2   = (const float*)d_in[13];
    const float* Wp1   = (const float*)d_in[14];
    const float* bp1   = (const float*)d_in[15];
    const float* Wp2   = (const float*)d_in[16];
    const float* bp2   = (const float*)d_in[17];
    const float* alpha = (const float*)d_in[18];

    float* ws       = (float*)d_ws;
    float* partials = ws;                 // 5 * 1024
    float* h        = ws + 5 * 1024;      // 2048
    float* gp       = h + D_DIM;          // 128

    igc_reduce<<<1024, 256, 0, stream>>>((const v4f*)x, (const v4f*)out,
                                         (const v4f*)noise, (const v4f*)imean,
                                         partials);
    igc_finalize<<<1, 256, 0, stream>>>(partials, usage, stats, gsig, rma, alpha,
                                        W1, b1, h);
    igc_gemv_wmma<<<16, 256, 0, stream>>>(h, Wg1, bg1, Wg2, Wp1, bp1, Wp2, gp);
    igc_out<<<1, 64, 0, stream>>>(gp, bg2, bp2, (float*)d_out);
}